// LinearCSD_56427280335527
// MI455X (gfx1250) — compile-verified
//
#include <hip/hip_runtime.h>
#include <cstdint>

// ---- problem dims (match reference) ----
constexpr int Bsz   = 4096;       // batch
constexpr int Dsz   = 16384;      // input size
constexpr int Kk    = 8;
constexpr int K1    = Kk + 1;     // 9 components
constexpr int Cc    = 2;          // classes
constexpr int NOUT  = K1 * Cc;    // 18 fused output columns (j = k*2 + c)

// ---- tiling ----
constexpr int KSPLIT    = 8;            // split-K segments over D
constexpr int KSEG      = Dsz / KSPLIT; // 2048
constexpr int KC        = 64;           // K-chunk staged in LDS
constexpr int NCHUNK    = KSEG / KC;    // 32
constexpr int MTILE     = 128;          // batch rows per workgroup (8 waves x 16)
constexpr int XS_STRIDE = 68;           // 64 + pad (272B rows, 16B aligned)
constexpr int WT_STRIDE = 34;           // 32 + pad (avoids 2-way conflicts)

typedef float v2f __attribute__((ext_vector_type(2)));
typedef float v8f __attribute__((ext_vector_type(8)));

// ---- async global->LDS path (gfx1250), guarded so compile can't regress ----
#if defined(__has_builtin)
#if __has_builtin(__builtin_amdgcn_global_load_async_to_lds_b128) && \
    __has_builtin(__builtin_amdgcn_global_load_async_to_lds_b64)
#define CSD_ASYNC_LDS 1
#endif
#endif
#ifndef CSD_ASYNC_LDS
#define CSD_ASYNC_LDS 0
#endif

#if CSD_ASYNC_LDS
// Param types per hipcc diagnostic: b128 wants int vector_size(16)*, b64 wants
// int vector_size(8)*; src is global AS, dst is LDS AS.
typedef int v4i __attribute__((vector_size(16)));
typedef int v2i __attribute__((vector_size(8)));
typedef __attribute__((address_space(1))) v4i glb_v4i;
typedef __attribute__((address_space(3))) v4i lds_v4i;
typedef __attribute__((address_space(1))) v2i glb_v2i;
typedef __attribute__((address_space(3))) v2i lds_v2i;

// Generic LDS pointer: low 32 bits are the LDS byte offset (ISA aperture rule),
// so an integer round-trip yields a valid AS(3) pointer value.
__device__ __forceinline__ glb_v4i* to_glb4(const void* p) {
  return (glb_v4i*)(uintptr_t)p;
}
__device__ __forceinline__ lds_v4i* to_lds4(const void* p) {
  return (lds_v4i*)(uint32_t)(uintptr_t)p;
}
__device__ __forceinline__ glb_v2i* to_glb2(const void* p) {
  return (glb_v2i*)(uintptr_t)p;
}
__device__ __forceinline__ lds_v2i* to_lds2(const void* p) {
  return (lds_v2i*)(uint32_t)(uintptr_t)p;
}
#endif

// proj partial GEMM: projp[ks][b][j] = sum_{d in seg ks} x[b,d] * W[d][j]
// W[d][j] = sms[k][d][c] with j = k*2 + c   (sms flat: (k*D + d)*2 + c)
__global__ __launch_bounds__(256) void csd_gemm_wmma(
    const float* __restrict__ x, const float* __restrict__ sms,
    float* __restrict__ projp) {
#if CSD_ASYNC_LDS
  __shared__ float xs[2][MTILE][XS_STRIDE];   // double-buffered
  __shared__ float wt[2][KC][WT_STRIDE];
#else
  __shared__ float xs[1][MTILE][XS_STRIDE];
  __shared__ float wt[1][KC][WT_STRIDE];
#endif

  const int tid  = threadIdx.x;
  const int wave = tid >> 5;     // 0..7
  const int lane = tid & 31;
  const int half = lane >> 4;    // K-pair select per ISA A/B f32 layout
  const int m16  = lane & 15;

  const int rowBase = blockIdx.x * MTILE;
  const int kBase   = blockIdx.y * KSEG;

  const int q  = tid & 15;   // float4 slot within 64-col chunk
  const int r0 = tid >> 4;   // 0..15
  const int dl = tid >> 2;   // 0..63: d-row for W staging
  const int k4 = tid & 3;

  // zero W tiles (pad cols NOUT..31 must stay 0 for the N=16..31 WMMA tile)
  for (int i = tid; i < (int)(sizeof(wt) / sizeof(float)); i += 256)
    (&wt[0][0][0])[i] = 0.0f;
  __syncthreads();

  v8f acc0 = {};  // N = 0..15
  v8f acc1 = {};  // N = 16..31 (only 16,17 real)

#if CSD_ASYNC_LDS
  // ---------------- async, double-buffered pipeline ----------------
  auto stage = [&](int chunk, int buf) {
    const int kcol = kBase + chunk * KC;
#pragma unroll
    for (int p = 0; p < 8; ++p) {
      const int r = p * 16 + r0;
      __builtin_amdgcn_global_load_async_to_lds_b128(
          to_glb4(&x[(size_t)(rowBase + r) * Dsz + kcol + q * 4]),
          to_lds4(&xs[buf][r][q * 4]), 0, 0);
    }
    const int d = kcol + dl;
#pragma unroll
    for (int kk = 0; kk < 2; ++kk) {
      const int k = k4 + kk * 4;
      __builtin_amdgcn_global_load_async_to_lds_b64(
          to_glb2(&sms[((size_t)k * Dsz + d) * Cc]),
          to_lds2(&wt[buf][dl][k * 2]), 0, 0);
    }
    if (k4 == 0)
      __builtin_amdgcn_global_load_async_to_lds_b64(
          to_glb2(&sms[((size_t)8 * Dsz + d) * Cc]),
          to_lds2(&wt[buf][dl][16]), 0, 0);
  };

  stage(0, 0);
  asm volatile("s_wait_asynccnt 0x0" ::: "memory");
  __syncthreads();

  for (int i = 0; i < NCHUNK; ++i) {
    const int cur = i & 1;
    if (i + 1 < NCHUNK) stage(i + 1, cur ^ 1);  // overlap with compute below

    const float* xrow = &xs[cur][wave * 16 + m16][0];
#pragma unroll
    for (int kk = 0; kk < KC; kk += 4) {
      v2f a, b0, b1;
      a.x  = xrow[kk + 2 * half + 0];
      a.y  = xrow[kk + 2 * half + 1];
      b0.x = wt[cur][kk + 2 * half + 0][m16];
      b0.y = wt[cur][kk + 2 * half + 1][m16];
      b1.x = wt[cur][kk + 2 * half + 0][16 + m16];
      b1.y = wt[cur][kk + 2 * half + 1][16 + m16];
      acc0 = __builtin_amdgcn_wmma_f32_16x16x4_f32(false, a, false, b0,
                                                   (short)0, acc0, false, false);
      acc1 = __builtin_amdgcn_wmma_f32_16x16x4_f32(false, a, false, b1,
                                                   (short)0, acc1, false, false);
    }
    asm volatile("s_wait_asynccnt 0x0" ::: "memory");
    __syncthreads();
  }
#else
  // ---------------- synchronous fallback (known-good) ----------------
  for (int kc = 0; kc < KSEG; kc += KC) {
    const int kcol = kBase + kc;
#pragma unroll
    for (int p = 0; p < 8; ++p) {
      const int r = p * 16 + r0;
      const float4 v = *reinterpret_cast<const float4*>(
          &x[(size_t)(rowBase + r) * Dsz + kcol + q * 4]);
      *reinterpret_cast<float4*>(&xs[0][r][q * 4]) = v;
    }
    {
      const int d = kcol + dl;
#pragma unroll
      for (int kk = 0; kk < 2; ++kk) {
        const int k = k4 + kk * 4;
        const float2 v = *reinterpret_cast<const float2*>(
            &sms[((size_t)k * Dsz + d) * Cc]);
        wt[0][dl][k * 2 + 0] = v.x;
        wt[0][dl][k * 2 + 1] = v.y;
      }
      if (k4 == 0) {
        const float2 v = *reinterpret_cast<const float2*>(
            &sms[((size_t)8 * Dsz + d) * Cc]);
        wt[0][dl][16] = v.x;
        wt[0][dl][17] = v.y;
      }
    }
    __syncthreads();

    const float* xrow = &xs[0][wave * 16 + m16][0];
#pragma unroll
    for (int kk = 0; kk < KC; kk += 4) {
      v2f a, b0, b1;
      a.x  = xrow[kk + 2 * half + 0];
      a.y  = xrow[kk + 2 * half + 1];
      b0.x = wt[0][kk + 2 * half + 0][m16];
      b0.y = wt[0][kk + 2 * half + 1][m16];
      b1.x = wt[0][kk + 2 * half + 0][16 + m16];
      b1.y = wt[0][kk + 2 * half + 1][16 + m16];
      acc0 = __builtin_amdgcn_wmma_f32_16x16x4_f32(false, a, false, b0,
                                                   (short)0, acc0, false, false);
      acc1 = __builtin_amdgcn_wmma_f32_16x16x4_f32(false, a, false, b1,
                                                   (short)0, acc1, false, false);
    }
    __syncthreads();
  }
#endif

  // write partial proj, layout [KSPLIT][B][NOUT]
  // C/D layout: VGPR v, lanes 0-15: M=v, N=lane; lanes 16-31: M=v+8, N=lane-16
  float* base = projp + (size_t)blockIdx.y * Bsz * NOUT;
#pragma unroll
  for (int v = 0; v < 8; ++v) {
    const int grow = rowBase + wave * 16 + v + 8 * half;
    base[(size_t)grow * NOUT + m16] = acc0[v];
    if (m16 < Cc)
      base[(size_t)grow * NOUT + 16 + m16] = acc1[v];
  }
}

// reduce split-K partials + compute both logit outputs
__global__ __launch_bounds__(256) void csd_epilogue(
    const float* __restrict__ projp, const int* __restrict__ domains,
    const float* __restrict__ smb, const float* __restrict__ embs,
    const float* __restrict__ cs_wt, float* __restrict__ out) {
  const int b = blockIdx.x * blockDim.x + threadIdx.x;
  if (b >= Bsz) return;

  float p[NOUT];
#pragma unroll
  for (int j = 0; j < NOUT; ++j) p[j] = 0.0f;
  for (int ks = 0; ks < KSPLIT; ++ks) {
    const float* src = projp + ((size_t)ks * Bsz + b) * NOUT;
#pragma unroll
    for (int j = 0; j < NOUT; ++j) p[j] += src[j];
  }

  float cw[K1];
  cw[0] = tanhf(cs_wt[0]);
  const int dom = domains[b];
#pragma unroll
  for (int k = 1; k < K1; ++k) cw[k] = tanhf(embs[dom * Kk + (k - 1)]);

#pragma unroll
  for (int c = 0; c < Cc; ++c) {
    out[(size_t)Bsz * Cc + (size_t)b * Cc + c] = p[c] + smb[c];  // common
    float s = 0.0f;
#pragma unroll
    for (int k = 0; k < K1; ++k)
      s += cw[k] * (p[k * Cc + c] + smb[k * Cc + c]);
    out[(size_t)b * Cc + c] = s;  // specialized
  }
}

// per-(c,k,j) gram entry -> ((cps - eye)^2)/162 term
__global__ __launch_bounds__(256) void csd_gram(
    const float* __restrict__ sms, float* __restrict__ cps_sq) {
  const int idx = blockIdx.x;          // 0..161
  const int c = idx / (K1 * K1);
  const int r = idx % (K1 * K1);
  const int k = r / K1;
  const int j = r % K1;
  const int tid = threadIdx.x;

  float s = 0.0f;
  for (int d = tid; d < Dsz; d += 256)
    s += sms[((size_t)k * Dsz + d) * Cc + c] *
         sms[((size_t)j * Dsz + d) * Cc + c];

  __shared__ float red[256];
  red[tid] = s;
  __syncthreads();
  for (int off = 128; off > 0; off >>= 1) {
    if (tid < off) red[tid] += red[tid + off];
    __syncthreads();
  }
  if (tid == 0) {
    const float diff = red[0] - (k == j ? 1.0f : 0.0f);
    cps_sq[idx] = diff * diff * (1.0f / (Cc * K1 * K1));
  }
}

__global__ __launch_bounds__(256) void csd_orth_final(
    const float* __restrict__ cps_sq, float* __restrict__ out) {
  const int tid = threadIdx.x;
  __shared__ float red[256];
  red[tid] = (tid < Cc * K1 * K1) ? cps_sq[tid] : 0.0f;
  __syncthreads();
  for (int off = 128; off > 0; off >>= 1) {
    if (tid < off) red[tid] += red[tid + off];
    __syncthreads();
  }
  if (tid == 0) out[(size_t)2 * Bsz * Cc] = red[0];
}

extern "C" void kernel_launch(void* const* d_in, const int* in_sizes, int n_in,
                              void* d_out, int out_size, void* d_ws, size_t ws_size,
                              hipStream_t stream) {
  const float* x       = (const float*)d_in[0];
  const int*   domains = (const int*)d_in[1];
  const float* sms     = (const float*)d_in[2];
  const float* smb     = (const float*)d_in[3];
  const float* embs    = (const float*)d_in[4];
  const float* cs_wt   = (const float*)d_in[5];
  float* out = (float*)d_out;

  float* projp  = (float*)d_ws;                         // KSPLIT*B*NOUT f32 (~2.36 MB)
  float* cps_sq = projp + (size_t)KSPLIT * Bsz * NOUT;  // 162 f32

  csd_gemm_wmma<<<dim3(Bsz / MTILE, KSPLIT), 256, 0, stream>>>(x, sms, projp);
  csd_epilogue<<<Bsz / 256, 256, 0, stream>>>(projp, domains, smb, embs, cs_wt, out);
  csd_gram<<<Cc * K1 * K1, 256, 0, stream>>>(sms, cps_sq);
  csd_orth_final<<<1, 256, 0, stream>>>(cps_sq, out);

  (void)in_sizes; (void)n_in; (void)out_size; (void)ws_size;
}